// InteractiveGallingModelV6_65111704207992
// MI455X (gfx1250) — compile-verified
//
#include <hip/hip_runtime.h>

// Reference: batched nonlinear stochastic scan (galling model).
// HBM-bound: ~354 MB moved total -> ~15 us floor at 23.3 TB/s.
// No matrix structure (sequential nonlinear recurrence) -> WMMA inapplicable.
// Strategy: streaming kernel, NT loads/stores, global_prefetch_b8, software
// pipelined loads, 2 chains per lane (float2), wave32-friendly 256-thread blocks.

typedef float v2f __attribute__((ext_vector_type(2)));

#define NCYC   150
#define BATCH  65536
#define HALF_B (BATCH / 2)
#define PF_DIST 8

__device__ __forceinline__ float sigmoid_f(float x) {
    // 1 / (1 + exp(-x)) using v_exp_f32 + v_rcp_f32 (TRANS pipe)
    return __builtin_amdgcn_rcpf(1.0f + __expf(-x));
}

__device__ __forceinline__ float softplus_f(float x) {
    // numerically stable: max(x,0) + log(1 + exp(-|x|))
    return fmaxf(x, 0.0f) + __logf(1.0f + __expf(-fabsf(x)));
}

__global__ __launch_bounds__(256) void galling_scan_kernel(
    const float* __restrict__ params,
    const float* __restrict__ Tptr,
    const float* __restrict__ u,
    const float* __restrict__ noise,
    float* __restrict__ out)
{
    const int tid = blockIdx.x * blockDim.x + threadIdx.x;   // 0 .. HALF_B-1
    const long long b  = 2LL * tid;
    const long long NB = (long long)NCYC * BATCH;

    // ---- hoist all t-invariant parameter algebra out of the scan ----
    const float dT    = Tptr[0] - 160.0f;                    // T - T_REF
    const float A     = params[0]  + params[1]  * dT;        // a0 + a_T*dT
    const float a_mu  = params[2];
    const float a_mu2 = params[3];
    const float C     = params[4]  + params[6]  * dT;        // c0 + c_T*dT
    const float c_mu  = params[5];
    const float S     = params[7]  + params[9]  * dT;        // s0 + s_T*dT
    const float s_mu  = params[8];
    const float J     = params[10] + params[12] * dT;        // j0 + j_T*dT
    const float j_mu  = params[11];
    const float v0    = params[13];
    const float v_mu  = params[14];
    float mu0 = params[15] + params[16] * dT;                // mu0_base + mu0_T*dT
    mu0 = fminf(fmaxf(mu0, 0.1f), 1.3f);

    const v2f* u2 = (const v2f*)u;      // index: t*HALF_B + tid
    const v2f* n2 = (const v2f*)noise;

    // per-channel output bases; channel order matches reference stack:
    // [mu, component, pi, d1, sigma1, d2, sigma2]
    float* o_mu = out + 0 * NB + b;
    float* o_cp = out + 1 * NB + b;
    float* o_pi = out + 2 * NB + b;
    float* o_d1 = out + 3 * NB + b;
    float* o_s1 = out + 4 * NB + b;
    float* o_d2 = out + 5 * NB + b;
    float* o_s2 = out + 6 * NB + b;

    v2f mu;
    mu[0] = mu0;
    mu[1] = mu0;

    // software pipeline: pre-issue loads for t=0
    v2f u_cur = __builtin_nontemporal_load(&u2[tid]);
    v2f n_cur = __builtin_nontemporal_load(&n2[tid]);

    for (int t = 0; t < NCYC; ++t) {
        // pull future lines toward L2 early (gfx1250 global_prefetch_b8)
        if (t + PF_DIST < NCYC) {
            __builtin_prefetch((const void*)&u2[(long long)(t + PF_DIST) * HALF_B + tid], 0, 1);
            __builtin_prefetch((const void*)&n2[(long long)(t + PF_DIST) * HALF_B + tid], 0, 1);
        }

        // issue next iteration's loads before computing (hide HBM latency
        // behind the serial mu dependency chain)
        v2f u_nxt = u_cur, n_nxt = n_cur;
        if (t + 1 < NCYC) {
            u_nxt = __builtin_nontemporal_load(&u2[(long long)(t + 1) * HALF_B + tid]);
            n_nxt = __builtin_nontemporal_load(&n2[(long long)(t + 1) * HALF_B + tid]);
        }

        v2f pi_v, d1_v, s1_v, d2_v, s2_v, cp_v, mun_v;
        #pragma unroll
        for (int k = 0; k < 2; ++k) {
            const float m  = mu[k];
            const float p  = sigmoid_f(fmaf(a_mu2, m * m, fmaf(a_mu, m, A)));
            const float d1 = fmaf(c_mu, m, C);
            const float s1 = softplus_f(fmaf(s_mu, m, S));
            const float d2 = fmaf(j_mu, m, J);
            const float s2 = softplus_f(fmaf(v_mu, m, v0));
            const bool  stay  = u_cur[k] < p;
            const float delta = stay ? fmaf(s1, n_cur[k], d1)
                                     : fmaf(s2, n_cur[k], d2);
            const float mn = fminf(fmaxf(m + delta, 0.1f), 1.3f);
            pi_v[k]  = p;
            d1_v[k]  = d1;
            s1_v[k]  = s1;
            d2_v[k]  = d2;
            s2_v[k]  = s2;
            cp_v[k]  = stay ? 0.0f : 1.0f;
            mun_v[k] = mn;
        }

        // streaming writes: 275 MB total > 192 MB L2, keep them non-temporal
        const long long row = (long long)t * BATCH;
        __builtin_nontemporal_store(mun_v, (v2f*)(o_mu + row));
        __builtin_nontemporal_store(cp_v,  (v2f*)(o_cp + row));
        __builtin_nontemporal_store(pi_v,  (v2f*)(o_pi + row));
        __builtin_nontemporal_store(d1_v,  (v2f*)(o_d1 + row));
        __builtin_nontemporal_store(s1_v,  (v2f*)(o_s1 + row));
        __builtin_nontemporal_store(d2_v,  (v2f*)(o_d2 + row));
        __builtin_nontemporal_store(s2_v,  (v2f*)(o_s2 + row));

        mu    = mun_v;
        u_cur = u_nxt;
        n_cur = n_nxt;
    }
}

extern "C" void kernel_launch(void* const* d_in, const int* in_sizes, int n_in,
                              void* d_out, int out_size, void* d_ws, size_t ws_size,
                              hipStream_t stream) {
    (void)in_sizes; (void)n_in; (void)out_size; (void)d_ws; (void)ws_size;
    const float* params = (const float*)d_in[0];   // 17 floats
    const float* T      = (const float*)d_in[1];   // scalar
    const float* u      = (const float*)d_in[2];   // [150, 65536]
    const float* noise  = (const float*)d_in[3];   // [150, 65536]
    float* out          = (float*)d_out;           // [7, 150, 65536]

    dim3 block(256);
    dim3 grid(HALF_B / 256);                       // 32768 threads = 1024 wave32
    hipLaunchKernelGGL(galling_scan_kernel, grid, block, 0, stream,
                       params, T, u, noise, out);
}